// MMCL_pgd_37778532335647
// MI455X (gfx1250) — compile-verified
//
#include <hip/hip_runtime.h>
#include <math.h>

// CDNA5 / gfx1250: wave32, WMMA (not MFMA), 320KB LDS per WGP.
typedef float v2f __attribute__((ext_vector_type(2)));
typedef float v8f __attribute__((ext_vector_type(8)));

#define BSZ   256      // batch size
#define NBLK  16       // workgroups; each owns BSZ/NBLK = 16 batch columns end-to-end
#define COLS  16       // columns (batch elements) per workgroup
#define PAD   260      // padded LDS row stride (floats) for KK: 260 % 64 == 4 -> conflict-free A-fragment ds_load_b64
#define NITER 300
#define ETA   0.001f

// ---------------- LDS layout (floats) ----------------
// KKl : 256*PAD   = 66560   (full 256x256 KK, padded rows)
// Zl  : 256*16    = 4096    (momentum iterate z, column-strided by 16)
// aL  : 256*16    = 4096
// apL : 256*16    = 4096
// Sg  : 16                  (column sums of Z)
// Gd  : 16                  (diag of G = KK@Z restricted to this block's columns)
// red : 256                 (reduction scratch)
#define OFF_KK  0
#define OFF_Z   (256*PAD)
#define OFF_A   (OFF_Z + 4096)
#define OFF_AP  (OFF_A + 4096)
#define OFF_S   (OFF_AP + 4096)
#define OFF_GD  (OFF_S + 16)
#define OFF_RED (OFF_GD + 16)
#define LDS_FLOATS (OFF_RED + 256)

// ---------------------------------------------------------------------------
// Phase 1: KK[x][j] = exp(-g*(2-2*A_x.A_j)), KsT[x][j] = Ks[j][x] = exp(-g*(2-2*A_j.B_x))
// features layout: (256, 2, 128) row-major; A_i = feat+i*256, B_i = feat+i*256+128
// ---------------------------------------------------------------------------
__global__ __launch_bounds__(256)
void rbf_build_kernel(const float* __restrict__ feat,
                      float* __restrict__ KK, float* __restrict__ KsT) {
  __shared__ float Ax[128];
  __shared__ float Bx[128];
  const int x = blockIdx.x;
  const int j = threadIdx.x;
  if (j < 128) Ax[j] = feat[x * 256 + j];
  else         Bx[j - 128] = feat[x * 256 + 128 + (j - 128)];
  __syncthreads();
  const float gamma = (float)(1.0 / 0.07);
  const float* Aj = feat + j * 256;
  float d0 = 0.0f, d1 = 0.0f;
  #pragma unroll 8
  for (int d = 0; d < 128; ++d) {
    float a = Aj[d];
    d0 = fmaf(a, Ax[d], d0);
    d1 = fmaf(a, Bx[d], d1);
  }
  KK [x * 256 + j] = expf(-gamma * (2.0f - 2.0f * d0));
  KsT[x * 256 + j] = expf(-gamma * (2.0f - 2.0f * d1));
}

// ---------------------------------------------------------------------------
// Phase 2: persistent per-WGP FISTA. Block b owns batch columns i = 16b..16b+15.
// grad[m,i] = S_i*(1-KK[m,i]) + 0.1*Z[m,i] + G[m,i] - G[i,i] - 1,  G = KK @ Z
// ---------------------------------------------------------------------------
__global__ __launch_bounds__(256)
void pgd_persistent_kernel(const float* __restrict__ KK,
                           const float* __restrict__ alpha_init,
                           float* __restrict__ alphab) {
  extern __shared__ float smem[];
  float* KKl = smem + OFF_KK;
  float* Zl  = smem + OFF_Z;
  float* aL  = smem + OFF_A;
  float* apL = smem + OFF_AP;
  float* Sg  = smem + OFF_S;
  float* Gd  = smem + OFF_GD;
  float* red = smem + OFF_RED;

  const int tid  = threadIdx.x;
  const int b    = blockIdx.x;        // 0..15
  const int lane = tid & 31;
  const int wave = tid >> 5;          // 0..7; wave owns M-tiles 2w and 2w+1

  // Stage KK (256KB) into LDS once via CDNA5 async global->LDS DMA:
  // per-lane 16B chunks, coalesced global reads, padded-row LDS destinations
  // (row byte stride 1040 = 65*16 and k%4==0 keep every dest 16B-aligned).
  // Tracked by ASYNCcnt, drained before the workgroup barrier.
  for (int e4 = tid; e4 < 16384; e4 += 256) {   // 16384 float4 chunks = 256KB
    int e = e4 << 2;
    int r = e >> 8, k = e & 255;
    unsigned lds_off = (unsigned)(((OFF_KK + r * PAD + k)) * 4);
    const float* gsrc = KK + e;
    asm volatile("global_load_async_to_lds_b128 %0, %1, off"
                 :: "v"(lds_off), "v"(gsrc) : "memory");
  }
  // alpha0 = clip(alpha_init, 0, 1) scattered to off-diagonal positions (diag=0).
  for (int e = tid; e < 4096; e += 256) {
    int c = e & 15, m = e >> 4;
    int i = b * COLS + c;
    float v = 0.0f;
    if (m != i) {
      int k = (m < i) ? m : (m - 1);
      v = fminf(fmaxf(alpha_init[i * 255 + k], 0.0f), 1.0f);
    }
    aL [m * 16 + c] = v;
    apL[m * 16 + c] = v;
  }
  asm volatile("s_wait_asynccnt 0x0" ::: "memory");
  __syncthreads();

  const int rA0 = (wave * 2 + 0) * 16 + (lane & 15);
  const int rA1 = (wave * 2 + 1) * 16 + (lane & 15);
  const int kof = (lane >> 4) * 2;     // per ISA A/B fragment layout (32-bit, 16x4 / 4x16)
  const int cB  = lane & 15;

  float t = 1.0f;
  for (int it = 0; it < NITER; ++it) {
    float tn   = (1.0f + sqrtf(1.0f + 4.0f * t * t)) * 0.5f;
    float beta = (t - 1.0f) / tn;
    t = tn;

    // z = a + beta*(a - a_prev), diag forced 0; per-column partial sums.
    {
      int c = tid & 15, mb = tid >> 4;
      int ig = b * COLS + c;
      float part = 0.0f;
      #pragma unroll 4
      for (int j = 0; j < 16; ++j) {
        int m = mb * 16 + j;
        int o = m * 16 + c;
        float a = aL[o], ap = apL[o];
        float z = a + beta * (a - ap);
        if (m == ig) z = 0.0f;
        Zl[o] = z;
        part += z;
      }
      red[tid] = part;
    }
    __syncthreads();
    if (tid < 128) red[tid] += red[tid + 128];
    __syncthreads();
    if (tid < 64)  red[tid] += red[tid + 64];
    __syncthreads();
    if (tid < 32)  red[tid] += red[tid + 32];
    __syncthreads();
    if (tid < 16)  { red[tid] += red[tid + 16]; Sg[tid] = red[tid]; }
    __syncthreads();

    // G = KK @ Z : 256x16 tile per block, two 16x16 tiles per wave, K = 256.
    // A fragments as 8B-aligned float2 -> ds_load_b64 / ds_load_2addr_b64.
    v8f acc0 = {}; v8f acc1 = {};
    #pragma unroll 8
    for (int k0 = 0; k0 < 256; k0 += 4) {
      int kk = k0 + kof;
      v2f a0 = *(const v2f*)&KKl[rA0 * PAD + kk];
      v2f a1 = *(const v2f*)&KKl[rA1 * PAD + kk];
      v2f bb = { Zl[kk * 16 + cB], Zl[(kk + 1) * 16 + cB] };
      acc0 = __builtin_amdgcn_wmma_f32_16x16x4_f32(false, a0, false, bb, (short)0, acc0, false, false);
      acc1 = __builtin_amdgcn_wmma_f32_16x16x4_f32(false, a1, false, bb, (short)0, acc1, false, false);
    }

    // Extract G[i,i] (lives in M-tile index b) into LDS for all waves.
    if ((b >> 1) == wave) {
      const v8f accd = (b & 1) ? acc1 : acc0;
      #pragma unroll
      for (int v = 0; v < 8; ++v) {
        int m = b * 16 + (lane >> 4) * 8 + v;
        int n = lane & 15;
        if (m == b * COLS + n) Gd[n] = accd[v];
      }
    }
    __syncthreads();

    // Proximal update straight from WMMA accumulators.
    #pragma unroll
    for (int half = 0; half < 2; ++half) {
      v8f acc = half ? acc1 : acc0;
      int tm = wave * 2 + half;
      #pragma unroll
      for (int v = 0; v < 8; ++v) {
        int m = tm * 16 + (lane >> 4) * 8 + v;
        int c = lane & 15;
        int i = b * COLS + c;
        int o = m * 16 + c;
        float z   = Zl[o];
        float g   = Sg[c] * (1.0f - KKl[m * PAD + i]) + 0.1f * z + acc[v] - Gd[c] - 1.0f;
        float an  = fminf(fmaxf(z - ETA * g, 0.0f), 1.0f);
        if (m == i) an = 0.0f;
        float aold = aL[o];
        apL[o] = aold;
        aL[o]  = an;
      }
    }
    __syncthreads();
  }

  // Write final alpha (column i contiguous in global, coalesced over m).
  for (int e = tid; e < 4096; e += 256) {
    int c = e >> 8, m = e & 255;
    alphab[(b * COLS + c) * 256 + m] = aL[m * 16 + c];
  }
}

// ---------------------------------------------------------------------------
// Phase 3: out = sum_t [ sum_m alpha[t][m]*KsT[t][m] - (sum_m alpha[t][m]) * KsT[t][t] / 256 ]
// (alpha diag is 0, so the m==t term of the first sum vanishes as in the reference)
// ---------------------------------------------------------------------------
__global__ __launch_bounds__(256)
void loss_reduce_kernel(const float* __restrict__ alphab,
                        const float* __restrict__ KsT,
                        float* __restrict__ out) {
  __shared__ float red[256];
  const int t = threadIdx.x;
  const float* ac = alphab + t * 256;
  const float* kc = KsT + t * 256;
  float neg = 0.0f, S = 0.0f;
  #pragma unroll 8
  for (int m = 0; m < 256; ++m) {
    float a = ac[m];
    neg = fmaf(a, kc[m], neg);
    S  += a;
  }
  red[t] = neg - S * kc[t] * (1.0f / 256.0f);
  __syncthreads();
  for (int s = 128; s > 0; s >>= 1) {
    if (t < s) red[t] += red[t + s];
    __syncthreads();
  }
  if (t == 0) out[0] = red[0];
}

// ---------------------------------------------------------------------------
extern "C" void kernel_launch(void* const* d_in, const int* in_sizes, int n_in,
                              void* d_out, int out_size, void* d_ws, size_t ws_size,
                              hipStream_t stream) {
  (void)in_sizes; (void)n_in; (void)out_size; (void)ws_size;
  const float* feat       = (const float*)d_in[0];  // (256,2,128) f32
  const float* alpha_init = (const float*)d_in[1];  // (256,255,1) f32

  float* KK     = (float*)d_ws;        // 256*256
  float* KsT    = KK  + 65536;         // 256*256 (Ks transposed)
  float* alphab = KsT + 65536;         // 256*256 final alpha, column-major per batch

  const size_t lds_bytes = (size_t)LDS_FLOATS * sizeof(float);  // ~309 KB < 320 KB/WGP
  hipFuncSetAttribute(reinterpret_cast<const void*>(pgd_persistent_kernel),
                      hipFuncAttributeMaxDynamicSharedMemorySize, (int)lds_bytes);

  rbf_build_kernel<<<BSZ, 256, 0, stream>>>(feat, KK, KsT);
  pgd_persistent_kernel<<<NBLK, 256, lds_bytes, stream>>>(KK, alpha_init, alphab);
  loss_reduce_kernel<<<1, 256, 0, stream>>>(alphab, KsT, (float*)d_out);
}